// GraphEncoder_51496657879183
// MI455X (gfx1250) — compile-verified
//
#include <hip/hip_runtime.h>
#include <math.h>

#define N_NODES 100000
#define OUT_CH  128
#define N_EDGES 600000
#define EPS_F   1e-10f
#define MB      5            // row-blocks (of 16) per wave; 6250 / 5 = 1250 exact

typedef __attribute__((ext_vector_type(2))) float v2f;
typedef __attribute__((ext_vector_type(8))) float v8f;

// ---------------------------------------------------------------------------
// Phase 0: zero the ptr accumulator in workspace (graph-capture safe).
// ---------------------------------------------------------------------------
__global__ void ge_zero_kernel(float* __restrict__ p, int n) {
    int i = blockIdx.x * blockDim.x + threadIdx.x;
    if (i < n) p[i] = 0.0f;
}

// ---------------------------------------------------------------------------
// Phase 1: message[e] = emb[sidx[e]] * edge_norm[e]; ptr = segment_sum(tidx).
// One thread per (edge, 4-channel group): float4 gather + 4 f32 atomics.
// ptr (51.2MB) fits in the 192MB L2 so the 76.8M RMWs stay on-chip.
// ---------------------------------------------------------------------------
__global__ void ge_scatter_kernel(const float* __restrict__ emb,
                                  const int*   __restrict__ edge_index,
                                  const float* __restrict__ edge_norm,
                                  float*       __restrict__ ptr_acc) {
    long long tid = (long long)blockIdx.x * blockDim.x + threadIdx.x;
    int e = (int)(tid >> 5);   // edge id
    int g = (int)(tid & 31);   // 4-channel group (32 groups * 4 = 128 ch)
    if (e >= N_EDGES) return;

    int   s = edge_index[e];            // edge_index[0][e]
    int   t = edge_index[N_EDGES + e];  // edge_index[1][e]
    float w = edge_norm[e];

    const float4 m = *((const float4*)(emb + (long long)s * OUT_CH) + g);
    float* dst = ptr_acc + (long long)t * OUT_CH + g * 4;
    atomicAdd(dst + 0, m.x * w);
    atomicAdd(dst + 1, m.y * w);
    atomicAdd(dst + 2, m.z * w);
    atomicAdd(dst + 3, m.w * w);
}

// ---------------------------------------------------------------------------
// Phase 2: loc = ptr @ loc_w.T + loc_b ; std = softplus(ptr @ std_w.T + std_b)
// fp32 matrix core: V_WMMA_F32_16X16X4_F32. Each wave owns MB=5 row-blocks
// (M=80) x one 16-col tile, for BOTH weight matrices: per K-step the two B
// fragments are loaded once and reused across 5 A fragments -> 7 loads per
// 10 WMMAs. The 5 A-row bases differ by a compile-time 8KB stride, folded
// into the load's 24-bit immediate offset (1 A pointer increment/iter, not 5).
//
// A (16x4, MxK):  lanes 0-15 -> K={0,1} of row M=lane; lanes 16-31 -> K={2,3}
// B (4x16, KxN):  columns across lanes; B[k][j] = W[j][k] -> contiguous
//                 float2 from row j of W per lane.
// C/D (16x16):    VGPR v -> row v + 8*(lane>=16), col = lane & 15.
// EXEC all-1s for every WMMA: grid divides N exactly (no tail, no guards).
// ---------------------------------------------------------------------------
__global__ void __launch_bounds__(32)
ge_gemm_kernel(const float* __restrict__ ptr_acc,
               const float* __restrict__ loc_w,
               const float* __restrict__ loc_b,
               const float* __restrict__ std_w,
               const float* __restrict__ std_b,
               float*       __restrict__ out) {
    const int lane = threadIdx.x;             // 0..31 (wave32)
    const int half = lane >> 4;               // 0: K-pair {0,1}, 1: {2,3}
    const int l    = lane & 15;
    const int m0   = blockIdx.x * (16 * MB);  // 1250 row groups
    const int j0   = blockIdx.y * 16;         // 8 col tiles

    const float* bLoc  = loc_w   + (long long)(j0 + l) * OUT_CH + half * 2;
    const float* bStd  = std_w   + (long long)(j0 + l) * OUT_CH + half * 2;
    const float* aBase = ptr_acc + (long long)(m0 + l) * OUT_CH + half * 2;

    v8f accL[MB];
    v8f accS[MB];
#pragma unroll
    for (int i = 0; i < MB; ++i) { accL[i] = (v8f){}; accS[i] = (v8f){}; }

#pragma unroll 4
    for (int kk = 0; kk < OUT_CH; kk += 4) {
        v2f bl = *(const v2f*)(bLoc + kk);
        v2f bs = *(const v2f*)(bStd + kk);
#pragma unroll
        for (int i = 0; i < MB; ++i) {
            // i * 16 rows * OUT_CH = const 2048 elements -> immediate offset
            v2f a = *(const v2f*)(aBase + i * (16 * OUT_CH) + kk);
            accL[i] = __builtin_amdgcn_wmma_f32_16x16x4_f32(
                false, a, false, bl, (short)0, accL[i], false, false);
            accS[i] = __builtin_amdgcn_wmma_f32_16x16x4_f32(
                false, a, false, bs, (short)0, accS[i], false, false);
        }
    }

    const int   col = j0 + l;
    const float lb  = loc_b[col];
    const float sb  = std_b[col];
    float* locOut = out + (long long)(m0 + half * 8) * OUT_CH + col;
    float* stdOut = locOut + (long long)N_NODES * OUT_CH;

#pragma unroll
    for (int i = 0; i < MB; ++i) {
#pragma unroll
        for (int v = 0; v < 8; ++v) {
            const int off = (16 * i + v) * OUT_CH;   // compile-time constant
            float lv = accL[i][v] + lb;
            float sv = accS[i][v] + sb;
            // fast softplus: 2 transcendentals; guard keeps large x exact and
            // avoids exp overflow (inputs here are ~O(0.1), guard never taken)
            sv = (sv > 80.0f) ? sv : __logf(1.0f + __expf(sv));
            locOut[off] = lv;
            stdOut[off] = sv + EPS_F;
        }
    }
}

// ---------------------------------------------------------------------------
// inputs (setup_inputs order): emb, loc_w, loc_b, std_w, std_b,
//                              edge_index, edge_norm
// out: loc (N*128 f32) then std (N*128 f32), concatenated flat.
// ws:  N*128 f32 accumulator (51.2 MB).
// ---------------------------------------------------------------------------
extern "C" void kernel_launch(void* const* d_in, const int* in_sizes, int n_in,
                              void* d_out, int out_size, void* d_ws, size_t ws_size,
                              hipStream_t stream) {
    const float* emb        = (const float*)d_in[0];
    const float* loc_w      = (const float*)d_in[1];
    const float* loc_b      = (const float*)d_in[2];
    const float* std_w      = (const float*)d_in[3];
    const float* std_b      = (const float*)d_in[4];
    const int*   edge_index = (const int*)  d_in[5];
    const float* edge_norm  = (const float*)d_in[6];
    float*       out        = (float*)d_out;
    float*       ptr_acc    = (float*)d_ws;

    const int nPtr = N_NODES * OUT_CH;
    ge_zero_kernel<<<(nPtr + 255) / 256, 256, 0, stream>>>(ptr_acc, nPtr);

    const long long scatterThreads = (long long)N_EDGES * 32;
    ge_scatter_kernel<<<(unsigned)((scatterThreads + 255) / 256), 256, 0, stream>>>(
        emb, edge_index, edge_norm, ptr_acc);

    dim3 grid(N_NODES / (16 * MB), OUT_CH / 16);   // 1250 x 8 waves
    ge_gemm_kernel<<<grid, 32, 0, stream>>>(ptr_acc, loc_w, loc_b,
                                            std_w, std_b, out);
}